// DiscriminatorNet_29781303230642
// MI455X (gfx1250) — compile-verified
//
#include <hip/hip_runtime.h>

typedef __attribute__((ext_vector_type(2))) float v2f;
typedef __attribute__((ext_vector_type(8))) float v8f;

#define N_REF   50000
#define BATCH   16
#define KDIM    4096          // 32 * 128
#define WAVES_PER_BLOCK 8
#define TILES   (N_REF / 16)  // 3125, exact

// -------------------------------------------------------------------------
// Kernel 1: for each 16-wide tile of n, compute dist[b][n] for all 16 batches
// using V_WMMA_F32_16X16X4_F32.  D = A(16x4, x) * B(4x16, raw^T) + C.
// r2[n] and x2[b] are accumulated from the same loaded elements (single pass).
// raw is streamed once (non-temporal); x stays hot in L2/WGP$.
// -------------------------------------------------------------------------
__global__ __launch_bounds__(256)
void dist_wmma_kernel(const float* __restrict__ x,
                      const float* __restrict__ raw,
                      float* __restrict__ dist /* [BATCH][N_REF] */) {
  const int lane = threadIdx.x & 31;
  const int wave = threadIdx.x >> 5;
  const int tile = blockIdx.x * WAVES_PER_BLOCK + wave;
  if (tile >= TILES) return;
  const int n0   = tile * 16;

  const int half = lane >> 4;   // 0: lanes 0-15 (K = +0,+1), 1: lanes 16-31 (K = +2,+3)
  const int lr   = lane & 15;

  // A operand (x):  lane holds x[M=lr][k + 2*half + {0,1}]   (16x4 f32 layout)
  const float* aptr = x   + (size_t)lr * KDIM + 2 * half;
  // B operand (raw^T): lane holds raw[n0+lr][k + 2*half + {0,1}]
  const float* bptr = raw + (size_t)(n0 + lr) * KDIM + 2 * half;

  v8f acc0 = {};
  v8f acc1 = {};
  float r2p = 0.0f;   // partial sum of squares of raw row n0+lr (this half's K)
  float x2p = 0.0f;   // partial sum of squares of x row lr (this half's K)

  for (int k = 0; k < KDIM; k += 8) {
    // prefetch the streaming operand ~1KB ahead; locality 3 -> SCOPE_WGP,
    // which is the only scope that pulls into all cache levels (ISA 10.5)
    __builtin_prefetch(bptr + k + 256, 0, 3);

    v2f a0 = *(const v2f*)(aptr + k);
    v2f b0 = __builtin_nontemporal_load((const v2f*)(bptr + k));
    v2f a1 = *(const v2f*)(aptr + k + 4);
    v2f b1 = __builtin_nontemporal_load((const v2f*)(bptr + k + 4));

    acc0 = __builtin_amdgcn_wmma_f32_16x16x4_f32(false, a0, false, b0,
                                                 (short)0, acc0, false, false);
    acc1 = __builtin_amdgcn_wmma_f32_16x16x4_f32(false, a1, false, b1,
                                                 (short)0, acc1, false, false);

    x2p += a0.x * a0.x + a0.y * a0.y + a1.x * a1.x + a1.y * a1.y;
    r2p += b0.x * b0.x + b0.y * b0.y + b1.x * b1.x + b1.y * b1.y;
  }

  v8f acc;
  #pragma unroll
  for (int j = 0; j < 8; ++j) acc[j] = acc0[j] + acc1[j];

  // Combine K-halves: lanes l and l+16 both end with the full row sums.
  r2p += __shfl_xor(r2p, 16, 32);   // r2[n0 + lr]
  x2p += __shfl_xor(x2p, 16, 32);   // x2[lr]

  // C/D layout: VGPR j, lane -> row M = j + 8*half, col N = lr.
  #pragma unroll
  for (int j = 0; j < 8; ++j) {
    const int b   = j + 8 * half;
    const float x2b = __shfl(x2p, b, 32);           // lane b holds x2[b]
    const float d   = r2p - 2.0f * acc[j] + x2b;
    dist[(size_t)b * N_REF + (n0 + lr)] = d;
  }
}

// -------------------------------------------------------------------------
// Kernel 2: per-batch row max and sum(exp(d - max)) over N_REF.
// -------------------------------------------------------------------------
__global__ __launch_bounds__(256)
void softmax_stats_kernel(const float* __restrict__ dist,
                          float* __restrict__ stats /* [BATCH][2] */) {
  const int b = blockIdx.x;
  const float* row = dist + (size_t)b * N_REF;
  __shared__ float red[256];

  float m = -3.402823466e38f;
  for (int i = threadIdx.x; i < N_REF; i += 256) m = fmaxf(m, row[i]);
  red[threadIdx.x] = m;
  __syncthreads();
  for (int s = 128; s > 0; s >>= 1) {
    if (threadIdx.x < s) red[threadIdx.x] = fmaxf(red[threadIdx.x], red[threadIdx.x + s]);
    __syncthreads();
  }
  const float rowmax = red[0];
  __syncthreads();

  float sum = 0.0f;
  for (int i = threadIdx.x; i < N_REF; i += 256) sum += expf(row[i] - rowmax);
  red[threadIdx.x] = sum;
  __syncthreads();
  for (int s = 128; s > 0; s >>= 1) {
    if (threadIdx.x < s) red[threadIdx.x] += red[threadIdx.x + s];
    __syncthreads();
  }
  if (threadIdx.x == 0) {
    stats[b * 2 + 0] = rowmax;
    stats[b * 2 + 1] = red[0];
  }
}

// -------------------------------------------------------------------------
// Kernel 3: out[b][n] = exp(dist - max) / sum
// -------------------------------------------------------------------------
__global__ __launch_bounds__(256)
void softmax_write_kernel(const float* __restrict__ dist,
                          const float* __restrict__ stats,
                          float* __restrict__ out) {
  const size_t i = (size_t)blockIdx.x * 256 + threadIdx.x;
  if (i >= (size_t)BATCH * N_REF) return;
  const int b = (int)(i / N_REF);
  const float m = stats[b * 2 + 0];
  const float s = stats[b * 2 + 1];
  out[i] = expf(dist[i] - m) / s;
}

extern "C" void kernel_launch(void* const* d_in, const int* in_sizes, int n_in,
                              void* d_out, int out_size, void* d_ws, size_t ws_size,
                              hipStream_t stream) {
  const float* x   = (const float*)d_in[0];   // [16, 32, 128]
  const float* raw = (const float*)d_in[1];   // [50000, 32, 128]
  float* out   = (float*)d_out;               // [16, 50000]
  float* dist  = (float*)d_ws;                // 16*50000 floats = 3.2 MB
  float* stats = dist + (size_t)BATCH * N_REF;

  const int blocks1 = (TILES + WAVES_PER_BLOCK - 1) / WAVES_PER_BLOCK;  // 391
  dist_wmma_kernel<<<blocks1, 256, 0, stream>>>(x, raw, dist);
  softmax_stats_kernel<<<BATCH, 256, 0, stream>>>(dist, stats);
  const int total = BATCH * N_REF;
  softmax_write_kernel<<<(total + 255) / 256, 256, 0, stream>>>(dist, stats, out);
}